// Affine_71081708749087
// MI455X (gfx1250) — compile-verified
//
#include <hip/hip_runtime.h>
#include <stdint.h>

// ---------------------------------------------------------------------------
// Affine grid_sample (bilinear, zeros padding, align_corners=True)
//   x[8,16,512,512] f32, theta[2,3] f32 -> out[8,16,512,512] f32
// Memory-bound: ~256 MiB total DRAM traffic -> ~11us at 23.3 TB/s.
// Grid is shared across all 128 planes -> compute taps once per block, reuse 16x.
// CDNA5 path: double-buffered GLOBAL_LOAD_ASYNC_TO_LDS_B128 staging of the
// tile's input footprint, sampled out of LDS. s_wait_asynccnt for sync.
// ---------------------------------------------------------------------------

#define AS1 __attribute__((address_space(1)))
#define AS3 __attribute__((address_space(3)))

typedef int v4i __attribute__((vector_size(16)));  // b128 payload type

#ifndef __has_builtin
#define __has_builtin(x) 0
#endif

#if __has_builtin(__builtin_amdgcn_global_load_async_to_lds_b128) && \
    __has_builtin(__builtin_amdgcn_s_wait_asynccnt)
#define ASYNC_OK 1
#else
#define ASYNC_OK 0
#endif

// Problem constants (from reference setup_inputs)
constexpr int BATCH = 8, CH = 16, H = 512, W = 512;
constexpr int NPLANE = BATCH * CH;   // 128 planes sharing one grid
constexpr int HW = H * W;

// Tiling
constexpr int TW = 64;                      // output tile width
constexpr int TH = 16;                      // output tile height
constexpr int PX = 4;                       // pixels per thread (TH/4 rows)
constexpr int TILES_X = W / TW;             // 8
constexpr int TILES_Y = H / TH;             // 32
constexpr int NTILES = TILES_X * TILES_Y;   // 256
constexpr int PLANES_PER_GROUP = 16;
constexpr int NGROUP = NPLANE / PLANES_PER_GROUP;  // 8

// LDS staging buffer (per plane footprint of one output tile)
constexpr int LW = 80;                      // floats per staged row
constexpr int LH = 24;                      // staged rows
constexpr int LVEC = LW / 4;                // 20 float4 per row
constexpr int KSTAGE = (LH * LVEC + 255) / 256;  // = 2 async b128 per thread

__global__ __launch_bounds__(256) void affine_sample_kernel(
    const float* __restrict__ x, const float* __restrict__ theta,
    float* __restrict__ out) {
  __shared__ float buf[2][LH * LW];
  __shared__ float dummy[4];

  const int tid  = threadIdx.x;
  const int tile = blockIdx.x;
  const int grp  = blockIdx.y;
  const int w0 = (tile % TILES_X) * TW;
  const int h0 = (tile / TILES_X) * TH;
  const int tx  = tid & (TW - 1);   // 0..63 -> output column in tile
  const int tyb = tid >> 6;         // 0..3  -> base row (stride 4)
  const int p0  = grp * PLANES_PER_GROUP;
  const int wpix = w0 + tx;

  const float t00 = theta[0], t01 = theta[1], t02 = theta[2];
  const float t10 = theta[3], t11 = theta[4], t12 = theta[5];
  const float stx = 2.0f / (float)(W - 1);
  const float sty = 2.0f / (float)(H - 1);
  const float cw  = 0.5f * (float)(W - 1);
  const float chh = 0.5f * (float)(H - 1);

  auto gridXY = [&](int w, int h, float& ix, float& iy) {
    float X = fmaf((float)w, stx, -1.0f);     // linspace(-1,1,W)[w]
    float Y = fmaf((float)h, sty, -1.0f);
    float gx = fmaf(t00, X, fmaf(t01, Y, t02));
    float gy = fmaf(t10, X, fmaf(t11, Y, t12));
    ix = (gx + 1.0f) * cw;                    // align_corners mapping
    iy = (gy + 1.0f) * chh;
  };

  // ---- input footprint of this tile (affine -> bbox at corners) ----------
  float ax0, ay0, ax1, ay1, ax2, ay2, ax3, ay3;
  gridXY(w0, h0, ax0, ay0);
  gridXY(w0 + TW - 1, h0, ax1, ay1);
  gridXY(w0, h0 + TH - 1, ax2, ay2);
  gridXY(w0 + TW - 1, h0 + TH - 1, ax3, ay3);
  const float ixmin = fminf(fminf(ax0, ax1), fminf(ax2, ax3));
  const float ixmax = fmaxf(fmaxf(ax0, ax1), fmaxf(ax2, ax3));
  const float iymin = fminf(fminf(ay0, ay1), fminf(ay2, ay3));
  const float iymax = fmaxf(fmaxf(ay0, ay1), fmaxf(ay2, ay3));
  int sx0g = max((int)floorf(ixmin) - 1, 0);            // -1 float safety
  int sx1g = min((int)floorf(ixmax) + 2, W - 1);        // +1 tap, +1 safety
  int sy0g = max((int)floorf(iymin) - 1, 0);
  int sy1g = min((int)floorf(iymax) + 2, H - 1);
  if (sx1g < sx0g) { sx0g = 0; sx1g = 0; }              // tile fully OOB
  if (sy1g < sy0g) { sy0g = 0; sy1g = 0; }
  const int sxa   = sx0g & ~3;                          // 16B-align left edge
  const int ncA4  = ((sx1g - sxa) >> 2) + 1;            // float4 per row
  const int nrows = sy1g - sy0g + 1;
  const bool staged = (ASYNC_OK != 0) && (ncA4 <= LVEC) && (nrows <= LH);

  // ---- per-pixel taps: weights (with zeros-padding masks) + offsets ------
  float w00[PX], w01[PX], w10[PX], w11[PX];
  int o00[PX], o01[PX], o10[PX], o11[PX];
#pragma unroll
  for (int i = 0; i < PX; ++i) {
    const int hpix = h0 + tyb + i * 4;
    float ix, iy;
    gridXY(wpix, hpix, ix, iy);
    const float xf = floorf(ix), yf = floorf(iy);
    const float fx = ix - xf, fy = iy - yf;
    const int xi = (int)xf, yi = (int)yf;
    const int xj = xi + 1,  yj = yi + 1;
    const float mx0 = (xi >= 0 && xi < W) ? 1.0f : 0.0f;
    const float mx1 = (xj >= 0 && xj < W) ? 1.0f : 0.0f;
    const float my0 = (yi >= 0 && yi < H) ? 1.0f : 0.0f;
    const float my1 = (yj >= 0 && yj < H) ? 1.0f : 0.0f;
    w00[i] = (1.0f - fx) * (1.0f - fy) * (mx0 * my0);
    w01[i] = fx * (1.0f - fy) * (mx1 * my0);
    w10[i] = (1.0f - fx) * fy * (mx0 * my1);
    w11[i] = fx * fy * (mx1 * my1);
    const int xc0 = min(max(xi, 0), W - 1), xc1 = min(max(xj, 0), W - 1);
    const int yc0 = min(max(yi, 0), H - 1), yc1 = min(max(yj, 0), H - 1);
    if (staged) {
      // LDS-local coords; clamp keeps in-bounds (only reachable w/ weight 0)
      const int lx0 = min(max(xc0 - sxa, 0), LW - 1);
      const int lx1 = min(max(xc1 - sxa, 0), LW - 1);
      const int ly0 = min(max(yc0 - sy0g, 0), LH - 1);
      const int ly1 = min(max(yc1 - sy0g, 0), LH - 1);
      o00[i] = ly0 * LW + lx0; o01[i] = ly0 * LW + lx1;
      o10[i] = ly1 * LW + lx0; o11[i] = ly1 * LW + lx1;
    } else {
      o00[i] = yc0 * W + xc0; o01[i] = yc0 * W + xc1;
      o10[i] = yc1 * W + xc0; o11[i] = yc1 * W + xc1;
    }
  }

#if ASYNC_OK
  if (staged) {
    // Precompute per-thread staging slots (plane-independent).
    const int totalVec = nrows * ncA4;
    const uint32_t ldsBase0 = (uint32_t)(uintptr_t)&buf[0][0];
    const uint32_t ldsBase1 = (uint32_t)(uintptr_t)&buf[1][0];
    const uint32_t ldsDummy = (uint32_t)(uintptr_t)&dummy[0];
    int gOff[KSTAGE];
    uint32_t lRel[KSTAGE];
    bool isDum[KSTAGE];
#pragma unroll
    for (int k = 0; k < KSTAGE; ++k) {
      const int v = tid + k * 256;
      if (v < totalVec) {
        const int r = v / ncA4;
        const int c = v - r * ncA4;
        gOff[k]  = (sy0g + r) * W + sxa + 4 * c;
        lRel[k]  = (uint32_t)(r * LW + 4 * c) * 4u;
        isDum[k] = false;
      } else {
        gOff[k]  = sy0g * W + sxa;  // harmless valid address
        lRel[k]  = 0;
        isDum[k] = true;            // park in dummy slot (keeps ASYNCcnt fixed)
      }
    }

    auto stage = [&](int plane, int bi) {
      const float* pb = x + (size_t)plane * HW;
      const uint32_t base = bi ? ldsBase1 : ldsBase0;
#pragma unroll
      for (int k = 0; k < KSTAGE; ++k) {
        const float* gsrc = pb + gOff[k];
        const uint32_t ldst = isDum[k] ? ldsDummy : (base + lRel[k]);
        __builtin_amdgcn_global_load_async_to_lds_b128(
            (AS1 v4i*)(uintptr_t)gsrc, (AS3 v4i*)(uintptr_t)ldst, 0, 0);
      }
    };

    stage(p0, 0);
    for (int pi = 0; pi < PLANES_PER_GROUP; ++pi) {
      const int p = p0 + pi;
      if (pi + 1 < PLANES_PER_GROUP) {
        stage(p + 1, (pi + 1) & 1);                     // prefetch next plane
        __builtin_amdgcn_s_wait_asynccnt(KSTAGE);       // current plane landed
      } else {
        __builtin_amdgcn_s_wait_asynccnt(0);
      }
      __syncthreads();                                  // all waves' DMA done
      const float* lb = &buf[pi & 1][0];
#pragma unroll
      for (int i = 0; i < PX; ++i) {
        float acc = lb[o00[i]] * w00[i];
        acc = fmaf(lb[o01[i]], w01[i], acc);
        acc = fmaf(lb[o10[i]], w10[i], acc);
        acc = fmaf(lb[o11[i]], w11[i], acc);
        out[(size_t)p * HW + (size_t)(h0 + tyb + i * 4) * W + wpix] = acc;
      }
      __syncthreads();  // reads done before buffer is overwritten next iter
    }
    return;
  }
#endif

  // ---- general fallback: direct gather (any theta / footprint) -----------
  for (int pi = 0; pi < PLANES_PER_GROUP; ++pi) {
    const int p = p0 + pi;
    const float* pb = x + (size_t)p * HW;
#pragma unroll
    for (int i = 0; i < PX; ++i) {
      float acc = pb[o00[i]] * w00[i];
      acc = fmaf(pb[o01[i]], w01[i], acc);
      acc = fmaf(pb[o10[i]], w10[i], acc);
      acc = fmaf(pb[o11[i]], w11[i], acc);
      out[(size_t)p * HW + (size_t)(h0 + tyb + i * 4) * W + wpix] = acc;
    }
  }
}

extern "C" void kernel_launch(void* const* d_in, const int* in_sizes, int n_in,
                              void* d_out, int out_size, void* d_ws,
                              size_t ws_size, hipStream_t stream) {
  (void)in_sizes; (void)n_in; (void)out_size; (void)d_ws; (void)ws_size;
  const float* x     = (const float*)d_in[0];
  const float* theta = (const float*)d_in[1];
  float* out         = (float*)d_out;
  dim3 grid(NTILES, NGROUP);  // 256 tiles x 8 plane-groups = 2048 blocks
  affine_sample_kernel<<<grid, 256, 0, stream>>>(x, theta, out);
}